// LINESG_63127429317269
// MI455X (gfx1250) — compile-verified
//
#include <hip/hip_runtime.h>
#include <hip/hip_bf16.h>

// ---------------------------------------------------------------------------
// LINE-SG pipeline for MI455X (gfx1250, wave32, WMMA)
//   cos = normalize(user_emb) @ normalize(user_emb)^T   (8192x8192, K=128)
//   ordered nonzero(cos > 0.3, excl diag + nonzero_idx) capped at K
//   gumbel-softmax head over E precomputed + K candidate pairs
// ---------------------------------------------------------------------------

#define N_USERS 8192
#define DIM     128
#define E_PAIRS 262144
#define K_CAND  131072
#define NTOT    (E_PAIRS + K_CAND)   // 393216
#define COS_THR 0.3f
#define TAU     0.2f
#define NTILE   (N_USERS / 16)       // 512 tiles per dimension
#define MASK_WORDS_PER_ROW (N_USERS / 32)  // 256

typedef __attribute__((ext_vector_type(2))) float v2f;
typedef __attribute__((ext_vector_type(8))) float v8f;

// ---------------------------------------------------------------------------
// One 16x16 fp32 cos tile via 32 x V_WMMA_F32_16X16X4_F32 (K = 128).
// A(16x4) lane L: m = L&15, ks = (L<16)?0:2  -> float2 u[i_base+m][kk+ks ..]
// B(4x16) lane L: n = L&15, ks = (L<16)?0:2  -> float2 u[j_base+n][kk+ks ..]
// C/D: VGPR r = rows (r | r+8), col = lane&15.
// ---------------------------------------------------------------------------
__device__ __forceinline__ v8f cos_tile_wmma(const float* __restrict__ u,
                                             int i_base, int j_base, int lane) {
  const int koff = (lane < 16) ? 0 : 2;
  const float* pa = u + (size_t)(i_base + (lane & 15)) * DIM + koff;
  const float* pb = u + (size_t)(j_base + (lane & 15)) * DIM + koff;
  v8f acc = {};
#pragma unroll
  for (int kk = 0; kk < DIM; kk += 4) {
    v2f a = *(const v2f*)(pa + kk);
    v2f b = *(const v2f*)(pb + kk);
    acc = __builtin_amdgcn_wmma_f32_16x16x4_f32(
        /*neg_a=*/false, a, /*neg_b=*/false, b,
        /*c_mod=*/(short)0, acc, /*reuse_a=*/false, /*reuse_b=*/false);
  }
  return acc;
}

// ---------------------------------------------------------------------------
// K0: zero exclusion bitmask + row counts, init candidate fill (pair=(0,0),
//     val=-1 == cos[0,0] after diag masking, matching jnp.nonzero fill).
// ---------------------------------------------------------------------------
__global__ void linesg_init(unsigned* __restrict__ exmask,
                            unsigned* __restrict__ counts,
                            int* __restrict__ cand_pairs,
                            float* __restrict__ cand_vals) {
  int idx = blockIdx.x * blockDim.x + threadIdx.x;
  int stride = gridDim.x * blockDim.x;
  const int mask_words = N_USERS * MASK_WORDS_PER_ROW;  // 2M words
  for (int t = idx; t < mask_words; t += stride) exmask[t] = 0u;
  for (int t = idx; t < N_USERS; t += stride) counts[t] = 0u;
  for (int t = idx; t < K_CAND; t += stride) {
    cand_vals[t] = -1.0f;
    cand_pairs[2 * t] = 0;
    cand_pairs[2 * t + 1] = 0;
  }
}

// K1: scatter exclusion bits (diagonal + directed nonzero_idx pairs)
__global__ void linesg_scatter(const int* __restrict__ nz,
                               unsigned* __restrict__ exmask) {
  int t = blockIdx.x * blockDim.x + threadIdx.x;
  if (t < N_USERS) {
    atomicOr(&exmask[t * MASK_WORDS_PER_ROW + (t >> 5)], 1u << (t & 31));
  }
  int p = t - N_USERS;
  if (p >= 0 && p < E_PAIRS) {
    int i = nz[2 * p], j = nz[2 * p + 1];
    atomicOr(&exmask[i * MASK_WORDS_PER_ROW + (j >> 5)], 1u << (j & 31));
  }
}

// K2: row-normalize embeddings (one wave32 per row, float4 per lane)
__global__ void linesg_normalize(const float* __restrict__ emb,
                                 float* __restrict__ u) {
  int lane = threadIdx.x & 31;
  int wave = threadIdx.x >> 5;
  int row = blockIdx.x * 8 + wave;
  const float4* src = (const float4*)(emb + (size_t)row * DIM);
  float4 v = src[lane];
  float ss = v.x * v.x + v.y * v.y + v.z * v.z + v.w * v.w;
#pragma unroll
  for (int off = 16; off; off >>= 1) ss += __shfl_xor(ss, off, 32);
  float inv = 1.0f / fmaxf(sqrtf(ss), 1e-12f);
  float4 o;
  o.x = v.x * inv; o.y = v.y * inv; o.z = v.z * inv; o.w = v.w * inv;
  ((float4*)(u + (size_t)row * DIM))[lane] = o;
}

// K3: count pass — one wave per 16x16 tile; per-row survivor counts
__global__ void linesg_count(const float* __restrict__ u,
                             const unsigned* __restrict__ exmask,
                             unsigned* __restrict__ counts) {
  const int lane = threadIdx.x & 31;
  const int wave = threadIdx.x >> 5;
  const int tile = blockIdx.x * 8 + wave;
  const int i_base = (tile >> 9) * 16;   // tile / 512
  const int j_base = (tile & 511) * 16;
  v8f acc = cos_tile_wmma(u, i_base, j_base, lane);
  const int j = j_base + (lane & 15);
#pragma unroll
  for (int r = 0; r < 8; ++r) {
    int i = i_base + r + ((lane < 16) ? 0 : 8);
    unsigned wbits = exmask[i * MASK_WORDS_PER_ROW + (j >> 5)];
    bool excl = (wbits >> (j & 31)) & 1u;
    bool pred = (acc[r] > COS_THR) && !excl;
    unsigned m = (unsigned)__ballot(pred);
    if (lane == 0)  atomicAdd(&counts[i_base + r],     (unsigned)__popc(m & 0xffffu));
    if (lane == 16) atomicAdd(&counts[i_base + r + 8], (unsigned)__popc(m >> 16));
  }
}

// K4: exclusive scan of 8192 row counts -> row_start (single workgroup)
__global__ void linesg_scan(const unsigned* __restrict__ counts,
                            unsigned* __restrict__ row_start) {
  __shared__ unsigned sSum[256];
  const int t = threadIdx.x;
  unsigned s = 0;
  for (int k = 0; k < 32; ++k) s += counts[t * 32 + k];
  sSum[t] = s;
  __syncthreads();
  for (int off = 1; off < 256; off <<= 1) {
    unsigned v = (t >= off) ? sSum[t - off] : 0u;
    __syncthreads();
    sSum[t] += v;
    __syncthreads();
  }
  unsigned run = (t == 0) ? 0u : sSum[t - 1];
  for (int k = 0; k < 32; ++k) {
    row_start[t * 32 + k] = run;
    run += counts[t * 32 + k];
  }
}

// K5: ordered emit — one workgroup per 16-row panel, 8 waves walk the 512
//     column tiles left-to-right (64 super-iterations). Ballot + LDS scan
//     gives exact row-major jnp.nonzero ordering; positions >= K dropped.
__global__ void linesg_emit(const float* __restrict__ u,
                            const unsigned* __restrict__ exmask,
                            const unsigned* __restrict__ row_start,
                            int* __restrict__ cand_pairs,
                            float* __restrict__ cand_vals) {
  __shared__ unsigned sCnt[16][8];
  __shared__ unsigned sBase[16][8];
  __shared__ unsigned sRow[16];
  const int lane = threadIdx.x & 31;
  const int wave = threadIdx.x >> 5;
  const int i_base = blockIdx.x * 16;
  if (wave == 0 && lane < 16) sRow[lane] = row_start[i_base + lane];
  __syncthreads();

  for (int super = 0; super < NTILE / 8; ++super) {
    const int j_base = (super * 8 + wave) * 16;
    v8f acc = cos_tile_wmma(u, i_base, j_base, lane);
    const int j = j_base + (lane & 15);
    unsigned ball[8];
#pragma unroll
    for (int r = 0; r < 8; ++r) {
      int i = i_base + r + ((lane < 16) ? 0 : 8);
      unsigned wbits = exmask[i * MASK_WORDS_PER_ROW + (j >> 5)];
      bool excl = (wbits >> (j & 31)) & 1u;
      bool pred = (acc[r] > COS_THR) && !excl;
      unsigned m = (unsigned)__ballot(pred);
      ball[r] = m;
      if (lane == 0)  sCnt[r][wave]     = (unsigned)__popc(m & 0xffffu);
      if (lane == 16) sCnt[r + 8][wave] = (unsigned)__popc(m >> 16);
    }
    __syncthreads();
    if (threadIdx.x < 16) {
      unsigned run = sRow[threadIdx.x];
#pragma unroll
      for (int w = 0; w < 8; ++w) {
        sBase[threadIdx.x][w] = run;
        run += sCnt[threadIdx.x][w];
      }
      sRow[threadIdx.x] = run;
    }
    __syncthreads();
#pragma unroll
    for (int r = 0; r < 8; ++r) {
      const bool lo = lane < 16;
      const int myrow = r + (lo ? 0 : 8);
      const unsigned sub = lo ? (ball[r] & 0xffffu) : (ball[r] >> 16);
      const int ln = lane & 15;
      if ((sub >> ln) & 1u) {
        unsigned pos = sBase[myrow][wave] + (unsigned)__popc(sub & ((1u << ln) - 1u));
        if (pos < (unsigned)K_CAND) {
          cand_pairs[2 * pos]     = i_base + myrow;
          cand_pairs[2 * pos + 1] = j_base + ln;
          cand_vals[pos]          = acc[r];
        }
      }
    }
    __syncthreads();
  }
}

// K6: gumbel-softmax head — one thread per pair; W staged in LDS
__global__ void linesg_gumbel(const float* __restrict__ emb,
                              const float* __restrict__ W,
                              const float* __restrict__ b,
                              const float* __restrict__ gn,
                              const int* __restrict__ nz,
                              const int* __restrict__ cand_pairs,
                              const float* __restrict__ cand_vals,
                              float* __restrict__ out) {
  __shared__ __align__(16) float sW[2 * 2 * DIM];  // 512
  __shared__ float sB[2];
  for (int t = threadIdx.x; t < 2 * 2 * DIM; t += blockDim.x) sW[t] = W[t];
  if (threadIdx.x < 2) sB[threadIdx.x] = b[threadIdx.x];
  __syncthreads();

  const int p = blockIdx.x * blockDim.x + threadIdx.x;
  if (p >= NTOT) return;

  int i, j;
  float weight;
  if (p < E_PAIRS) {
    i = nz[2 * p];
    j = nz[2 * p + 1];
    weight = 1.0f;
  } else {
    int q = p - E_PAIRS;
    i = cand_pairs[2 * q];
    j = cand_pairs[2 * q + 1];
    weight = fmaxf(cand_vals[q], 0.0f);
  }

  const float4* ei = (const float4*)(emb + (size_t)i * DIM);
  const float4* ej = (const float4*)(emb + (size_t)j * DIM);
  const float4* w4 = (const float4*)sW;  // [0]:W0[0:128] [32]:W0[128:256] [64]:W1[0:128] [96]:W1[128:256]
  float l0 = 0.0f, l1 = 0.0f;
#pragma unroll 8
  for (int d = 0; d < DIM / 4; ++d) {
    float4 e = ei[d];
    float4 a0 = w4[d], a1 = w4[64 + d];
    l0 += e.x * a0.x + e.y * a0.y + e.z * a0.z + e.w * a0.w;
    l1 += e.x * a1.x + e.y * a1.y + e.z * a1.z + e.w * a1.w;
    float4 f = ej[d];
    float4 b0 = w4[32 + d], b1 = w4[96 + d];
    l0 += f.x * b0.x + f.y * b0.y + f.z * b0.z + f.w * b0.w;
    l1 += f.x * b1.x + f.y * b1.y + f.z * b1.z + f.w * b1.w;
  }
  l0 += sB[0];
  l1 += sB[1];

  float z0 = (l0 + gn[2 * p]) / TAU;
  float z1 = (l1 + gn[2 * p + 1]) / TAU;
  float m = fmaxf(z0, z1);
  float e0 = expf(z0 - m), e1 = expf(z1 - m);
  float inv = 1.0f / (e0 + e1);
  float y0 = e0 * inv;
  float h0 = (z0 >= z1) ? 1.0f : 0.0f;   // argmax, first-index tie-break
  float r = (h0 + y0) - y0;              // y_hard + y_soft - stop_grad(y_soft)
  float w = r * weight;

  out[p] = w;                                   // gumbel_01
  out[NTOT + p] = w;                            // gumbel_retain_w[:NTOT]
  out[2 * NTOT + p] = w;                        // gumbel_retain_w[NTOT:]
  out[3 * NTOT + 2 * p] = (float)i;             // pair_idx
  out[3 * NTOT + 2 * p + 1] = (float)j;
}

// ---------------------------------------------------------------------------
extern "C" void kernel_launch(void* const* d_in, const int* in_sizes, int n_in,
                              void* d_out, int out_size, void* d_ws, size_t ws_size,
                              hipStream_t stream) {
  const float* user_emb = (const float*)d_in[0];
  const float* W        = (const float*)d_in[1];
  const float* b        = (const float*)d_in[2];
  const float* gn       = (const float*)d_in[3];
  const int*   nz       = (const int*)d_in[4];
  float* out = (float*)d_out;

  char* ws = (char*)d_ws;
  float*    u          = (float*)ws;                                  // 4 MB
  unsigned* exmask     = (unsigned*)(ws + (4u << 20));                // 8 MB
  unsigned* counts     = (unsigned*)(ws + (12u << 20));               // 32 KB
  unsigned* row_start  = (unsigned*)(ws + (12u << 20) + 32768u);      // 32 KB
  int*      cand_pairs = (int*)(ws + (12u << 20) + 65536u);           // 1 MB
  float*    cand_vals  = (float*)(ws + (13u << 20) + 65536u);         // 512 KB

  linesg_init<<<2048, 256, 0, stream>>>(exmask, counts, cand_pairs, cand_vals);
  linesg_scatter<<<(N_USERS + E_PAIRS + 255) / 256, 256, 0, stream>>>(nz, exmask);
  linesg_normalize<<<N_USERS / 8, 256, 0, stream>>>(user_emb, u);
  linesg_count<<<(NTILE * NTILE) / 8, 256, 0, stream>>>(u, exmask, counts);
  linesg_scan<<<1, 256, 0, stream>>>(counts, row_start);
  linesg_emit<<<NTILE, 256, 0, stream>>>(u, exmask, row_start, cand_pairs, cand_vals);
  linesg_gumbel<<<(NTOT + 255) / 256, 256, 0, stream>>>(user_emb, W, b, gn, nz,
                                                        cand_pairs, cand_vals, out);
}